// LSTMModel_82360292868323
// MI455X (gfx1250) — compile-verified
//
#include <hip/hip_runtime.h>
#include <math.h>

typedef __attribute__((ext_vector_type(2))) float v2f;
typedef __attribute__((ext_vector_type(8))) float v8f;

#define HD   1024
#define TT   128
#define G4   4096
#define RECB 128   // recurrence blocks (each owns 8 h-indices)
#define RECT 256   // threads per recurrence block

__device__ __forceinline__ float sigf(float x) { return 1.0f / (1.0f + __expf(-x)); }

// ---------------------------------------------------------------------------
// C[M,N] = A[M,K] @ W[N,K]^T + bias0[N] + bias1[N]   (fp32 WMMA 16x16x4)
// One wave per 16x16 C tile, 8 waves per block.
// ---------------------------------------------------------------------------
__global__ __launch_bounds__(256) void gemm_bias_wmma(
    const float* __restrict__ A, const float* __restrict__ W,
    const float* __restrict__ bias0, const float* __restrict__ bias1,
    float* __restrict__ C, int M, int N, int K)
{
    const int lane  = threadIdx.x & 31;
    const int wave  = threadIdx.x >> 5;
    const int tilesN = N >> 4;
    const int tile  = blockIdx.x * 8 + wave;
    const int tm = (tile / tilesN) << 4;
    const int tn = (tile % tilesN) << 4;
    const int half = lane >> 4;         // 0: K pair {0,1}; 1: K pair {2,3}
    const int l15  = lane & 15;

    const float* Ap = A + (size_t)(tm + l15) * K + 2 * half;
    const float* Wp = W + (size_t)(tn + l15) * K + 2 * half;

    v8f acc = {};
    for (int k0 = 0; k0 < K; k0 += 4) {
        v2f a = *(const v2f*)(Ap + k0);   // A[m][k0+2h], A[m][k0+2h+1]
        v2f b = *(const v2f*)(Wp + k0);   // B[k0+2h][n], B[k0+2h+1][n] == W[n][k...]
        acc = __builtin_amdgcn_wmma_f32_16x16x4_f32(
            false, a, false, b, (short)0, acc, false, false);
    }
    const float bias = bias0[tn + l15] + bias1[tn + l15];
#pragma unroll
    for (int v = 0; v < 8; ++v) {
        int m = tm + v + 8 * half;
        C[(size_t)m * N + tn + l15] = acc[v] + bias;
    }
}

// ---------------------------------------------------------------------------
// Persistent weight-stationary LSTM recurrence (batch size 1).
// Block b owns h-indices [b*8, b*8+8): gate rows {q*1024 + b*8 + jj}.
// 32 rows x 1024 fp32 weights pinned in LDS; per step only h (4KB) moves.
// Grid-wide sense barrier via monotone atomic counter (zeroed pre-launch).
// ---------------------------------------------------------------------------
__global__ __launch_bounds__(RECT) void lstm_rec(
    const float* __restrict__ xg,    // [T,4096] input projections + biases
    const float* __restrict__ Whh,   // [4096,1024]
    float* __restrict__ hseq,        // [T,1024] output
    float* hglob,                    // [1024] shared h (pre-zeroed)
    unsigned* ctr, int T)
{
    extern __shared__ float smem[];
    float* Wl   = smem;              // 32*1024
    float* hbuf = Wl + 32 * HD;      // 1024
    float* gbuf = hbuf + HD;         // 32
    float* cbuf = gbuf + 32;         // 8

    const int tid = threadIdx.x;
    const int j0  = blockIdx.x * 8;

    // Pin this block's 32 gate rows of Whh into LDS (float4 loads).
    for (int idx = tid; idx < 32 * (HD / 4); idx += RECT) {
        int lr = idx >> 8;                  // local row 0..31 (256 float4/row)
        int kq = idx & 255;
        int q = lr >> 3, jj = lr & 7;
        const float4* src = (const float4*)(Whh + (size_t)(q * HD + j0 + jj) * HD);
        ((float4*)(Wl + lr * HD))[kq] = src[kq];
    }
    if (tid < 8) cbuf[tid] = 0.0f;
    __syncthreads();

    const int lr   = tid >> 3;       // local gate row 0..31
    const int sub  = tid & 7;        // 8 threads per row
    const int q    = lr >> 3;
    const int jj   = lr & 7;
    const int grow = q * HD + j0 + jj;
    const float* wrow = Wl + lr * HD;

    for (int t = 0; t < T; ++t) {
        // Stage h (device-coherent loads: bypass stale WGP$)
        for (int i = tid; i < HD; i += RECT)
            hbuf[i] = __hip_atomic_load(hglob + i, __ATOMIC_RELAXED,
                                        __HIP_MEMORY_SCOPE_AGENT);
        __syncthreads();

        // 8-lane split dot product, strided to avoid LDS bank conflicts
        float s = 0.0f;
        for (int k = sub; k < HD; k += 8) s += wrow[k] * hbuf[k];
        s += __shfl_down(s, 4, 8);
        s += __shfl_down(s, 2, 8);
        s += __shfl_down(s, 1, 8);
        if (sub == 0) gbuf[lr] = s + xg[(size_t)t * G4 + grow];
        __syncthreads();

        // Pointwise gate update for the 8 owned h-indices
        if (tid < 8) {
            float gi = gbuf[0 * 8 + tid];
            float gf = gbuf[1 * 8 + tid];
            float gg = gbuf[2 * 8 + tid];
            float go = gbuf[3 * 8 + tid];
            float c  = sigf(gf) * cbuf[tid] + sigf(gi) * tanhf(gg);
            cbuf[tid] = c;
            float h = sigf(go) * tanhf(c);
            hseq[(size_t)t * HD + j0 + tid] = h;
            __hip_atomic_store(hglob + j0 + tid, h, __ATOMIC_RELAXED,
                               __HIP_MEMORY_SCOPE_AGENT);
        }
        __syncthreads();

        // Grid barrier: monotone counter, target grows by gridDim each step
        if (tid == 0) {
            __threadfence();
            __hip_atomic_fetch_add(ctr, 1u, __ATOMIC_ACQ_REL, __HIP_MEMORY_SCOPE_AGENT);
            unsigned target = (unsigned)(t + 1) * gridDim.x;
            while (__hip_atomic_load(ctr, __ATOMIC_ACQUIRE,
                                     __HIP_MEMORY_SCOPE_AGENT) < target)
                __builtin_amdgcn_s_sleep(1);
        }
        __syncthreads();
    }
}

// ---------------------------------------------------------------------------
// out[t] = dot(hseq2[t,:], Wlin) + blin   (128 dots of 1024)
// ---------------------------------------------------------------------------
__global__ __launch_bounds__(256) void final_linear(
    const float* __restrict__ hseq, const float* __restrict__ Wlin,
    const float* __restrict__ blin, float* __restrict__ out)
{
    __shared__ float red[256];
    const int t = blockIdx.x;
    float s = 0.0f;
    for (int k = threadIdx.x; k < HD; k += 256)
        s += hseq[(size_t)t * HD + k] * Wlin[k];
    red[threadIdx.x] = s;
    __syncthreads();
    for (int st = 128; st > 0; st >>= 1) {
        if (threadIdx.x < st) red[threadIdx.x] += red[threadIdx.x + st];
        __syncthreads();
    }
    if (threadIdx.x == 0) out[t] = red[0] + blin[0];
}

// ---------------------------------------------------------------------------
extern "C" void kernel_launch(void* const* d_in, const int* in_sizes, int n_in,
                              void* d_out, int out_size, void* d_ws, size_t ws_size,
                              hipStream_t stream) {
    const float* x    = (const float*)d_in[0];   // [128,128,512]
    const float* Wih0 = (const float*)d_in[1];   // [4096,512]
    const float* Whh0 = (const float*)d_in[2];   // [4096,1024]
    const float* bih0 = (const float*)d_in[3];
    const float* bhh0 = (const float*)d_in[4];
    const float* Wih1 = (const float*)d_in[5];   // [4096,1024]
    const float* Whh1 = (const float*)d_in[6];   // [4096,1024]
    const float* bih1 = (const float*)d_in[7];
    const float* bhh1 = (const float*)d_in[8];
    const float* Wlin = (const float*)d_in[9];   // [1,1024]
    const float* blin = (const float*)d_in[10];
    float* out = (float*)d_out;

    // Workspace layout (floats)
    float* xg    = (float*)d_ws;                 // 128*4096
    float* hseq1 = xg + (size_t)TT * G4;         // 128*1024
    float* hseq2 = hseq1 + (size_t)TT * HD;      // 128*1024
    float* hglob = hseq2 + (size_t)TT * HD;      // 1024
    unsigned* ctr = (unsigned*)(hglob + HD);     // barrier counter

    const size_t smem = (32 * HD + HD + 32 + 8) * sizeof(float);  // ~132 KB
    const int gemm_blocks = (TT / 16) * (G4 / 16) / 8;            // 256

    // ---- Layer 0: only batch element 127 matters (h2[-1] dependency) ----
    hipMemsetAsync(hglob, 0, HD * sizeof(float) + 64, stream);
    gemm_bias_wmma<<<gemm_blocks, 256, 0, stream>>>(
        x + (size_t)127 * TT * 512, Wih0, bih0, bhh0, xg, TT, G4, 512);
    lstm_rec<<<RECB, RECT, smem, stream>>>(xg, Whh0, hseq1, hglob, ctr, TT);

    // ---- Layer 1 ----
    hipMemsetAsync(hglob, 0, HD * sizeof(float) + 64, stream);
    gemm_bias_wmma<<<gemm_blocks, 256, 0, stream>>>(
        hseq1, Wih1, bih1, bhh1, xg, TT, G4, HD);
    lstm_rec<<<RECB, RECT, smem, stream>>>(xg, Whh1, hseq2, hglob, ctr, TT);

    // ---- out[t] = h2seq[t] . Wlin + b ----
    final_linear<<<TT, 256, 0, stream>>>(hseq2, Wlin, blin, out);
}